// QConv_2954937500420
// MI455X (gfx1250) — compile-verified
//
#include <hip/hip_runtime.h>
#include <math.h>

typedef float v2f __attribute__((ext_vector_type(2)));
typedef float v8f __attribute__((ext_vector_type(8)));

#define N_POSI 1021
#define OUT_STRIDE_B 4084   // 4 * 1021

// ---------------------------------------------------------------------------
// Kernel 1: build the per-position 16x16 complex unitary U_p (the 2 layers of
// Rot gates + CNOT-ring permutations). One thread per (p, column). The whole
// column (16 complex values) lives in registers via full unrolling.
// Output layout in ws: per p, 512 floats: [0..255] = Re(U) row-major 16x16,
// [256..511] = Im(U) row-major 16x16.
// ---------------------------------------------------------------------------
__global__ void build_unitaries_kernel(const float* __restrict__ params,
                                       float* __restrict__ U)
{
    int gid = blockIdx.x * blockDim.x + threadIdx.x;
    if (gid >= N_POSI * 16) return;
    const int p   = gid >> 4;
    const int col = gid & 15;

    float ur[16], ui[16];
#pragma unroll
    for (int d = 0; d < 16; ++d) { ur[d] = (d == col) ? 1.0f : 0.0f; ui[d] = 0.0f; }

    const float* pp = params + p * 24;   // (N_LAYERS=2, N_WIRES=4, 3)

#pragma unroll
    for (int l = 0; l < 2; ++l) {
        // --- Rot gates on each wire (wire w acts on bit (3-w)) ---
#pragma unroll
        for (int w = 0; w < 4; ++w) {
            const float phi   = pp[(l * 4 + w) * 3 + 0];
            const float theta = pp[(l * 4 + w) * 3 + 1];
            const float omega = pp[(l * 4 + w) * 3 + 2];
            float st, ct; __sincosf(0.5f * theta,          &st, &ct);
            float sp, cp; __sincosf(0.5f * (phi + omega),  &sp, &cp);
            float sm, cm; __sincosf(0.5f * (phi - omega),  &sm, &cm);
            // m00 = e^{-i(phi+omega)/2} c ; m01 = -e^{+i(phi-omega)/2} s
            // m10 = e^{-i(phi-omega)/2} s ; m11 = e^{+i(phi+omega)/2} c
            const float m00r =  cp * ct, m00i = -sp * ct;
            const float m01r = -cm * st, m01i = -sm * st;
            const float m10r =  cm * st, m10i = -sm * st;
            const float m11r =  cp * ct, m11i =  sp * ct;

            const int bit = 1 << (3 - w);
#pragma unroll
            for (int d0 = 0; d0 < 16; ++d0) {
                if (d0 & bit) continue;          // compile-time after unroll
                const int d1 = d0 | bit;
                const float ar = ur[d0], ai = ui[d0];
                const float br = ur[d1], bi = ui[d1];
                ur[d0] = m00r * ar - m00i * ai + m01r * br - m01i * bi;
                ui[d0] = m00r * ai + m00i * ar + m01r * bi + m01i * br;
                ur[d1] = m10r * ar - m10i * ai + m11r * br - m11i * bi;
                ui[d1] = m10r * ai + m10i * ar + m11r * bi + m11i * br;
            }
        }
        // --- CNOT ring: new_state[d] = state[perm[d]] (row permutation) ---
#pragma unroll
        for (int w = 0; w < 4; ++w) {
            const int pc = 3 - w;
            const int pt = 3 - ((w + 1) & 3);
            float nr[16], ni[16];
#pragma unroll
            for (int d = 0; d < 16; ++d) {
                const int src = d ^ (((d >> pc) & 1) << pt);
                nr[d] = ur[src]; ni[d] = ui[src];
            }
#pragma unroll
            for (int d = 0; d < 16; ++d) { ur[d] = nr[d]; ui[d] = ni[d]; }
        }
    }

    float* Up = U + p * 512;
#pragma unroll
    for (int d = 0; d < 16; ++d) {
        Up[d * 16 + col]       = ur[d];
        Up[256 + d * 16 + col] = ui[d];
    }
}

// ---------------------------------------------------------------------------
// Kernel 2: one wave32 per (position p, batch-tile of 16).
//   Yr = Re(U_p) * S0 ; Yi = Im(U_p) * S0   via 8x V_WMMA_F32_16X16X4_F32
//   probs = Yr^2 + Yi^2 ; out[b,c,p] = sum_d sign(c,d) * probs[d,b]
// ---------------------------------------------------------------------------
__global__ void __launch_bounds__(256)
qconv_wmma_kernel(const float* __restrict__ x,
                  const float* __restrict__ U,
                  float* __restrict__ out)
{
    const int lane = threadIdx.x & 31;
    const int wid  = blockIdx.x * (blockDim.x >> 5) + (threadIdx.x >> 5);
    const int p    = wid >> 5;          // 0..1020
    const int tile = wid & 31;          // 0..31
    if (p >= N_POSI) return;            // uniform per wave

    const bool hi = (lane >= 16);
    const int  n  = lane & 15;          // column within the tile
    const int  b  = tile * 16 + n;      // batch index this lane's column maps to

    // ---- product state for batch b at position p (wire w lives at bit 3-w) --
    const float* xr = x + b * 1024 + p;
    float c0, s0, c1, s1, c2, s2, c3, s3;
    __sincosf(0.5f * xr[0], &s0, &c0);
    __sincosf(0.5f * xr[1], &s1, &c1);
    __sincosf(0.5f * xr[2], &s2, &c2);
    __sincosf(0.5f * xr[3], &s3, &c3);

    float st[16];
#pragma unroll
    for (int d = 0; d < 16; ++d) {
        st[d] = (((d >> 3) & 1) ? s0 : c0) * (((d >> 2) & 1) ? s1 : c1) *
                (((d >> 1) & 1) ? s2 : c2) * (((d >> 0) & 1) ? s3 : c3);
    }

    // ---- WMMA operands -----------------------------------------------------
    // A (16x4 f32): lanes 0-15 -> row m, K = 4k+{0,1}; lanes 16-31 -> K = 4k+{2,3}
    // B (4x16 f32): VGPR0 lanes0-15 = row 4k+0, lanes16-31 = row 4k+2;
    //               VGPR1 lanes0-15 = row 4k+1, lanes16-31 = row 4k+3
    const float* Up  = U + p * 512;
    const int    m   = lane & 15;
    const int    koff = hi ? 2 : 0;

    v2f ar[4], ai[4], bb[4];
#pragma unroll
    for (int kk = 0; kk < 4; ++kk) {
        const float* pr = Up + m * 16 + kk * 4 + koff;   // 8-byte aligned pair
        ar[kk].x = pr[0];    ar[kk].y = pr[1];
        ai[kk].x = pr[256];  ai[kk].y = pr[257];
        bb[kk].x = hi ? st[kk * 4 + 2] : st[kk * 4 + 0];
        bb[kk].y = hi ? st[kk * 4 + 3] : st[kk * 4 + 1];
    }

    v8f cr = {}; v8f ci = {};
#pragma unroll
    for (int kk = 0; kk < 4; ++kk) {
        cr = __builtin_amdgcn_wmma_f32_16x16x4_f32(false, ar[kk], false, bb[kk],
                                                   (short)0, cr, false, false);
        ci = __builtin_amdgcn_wmma_f32_16x16x4_f32(false, ai[kk], false, bb[kk],
                                                   (short)0, ci, false, false);
    }

    // ---- epilogue: probs, signed Z reductions ------------------------------
    // D layout: lane col = n, VGPR v = row (v + (hi?8:0)).
    // sign(c,d) = 1 - 2*bit(3-c, d); c=0 -> bit3 (=hi), c=1 -> bit2 (=v>>2), ...
    float sum0 = 0.f, a1 = 0.f, a2 = 0.f, a3 = 0.f;
#pragma unroll
    for (int v = 0; v < 8; ++v) {
        const float pv = cr[v] * cr[v] + ci[v] * ci[v];
        sum0 += pv;
        a1 += ((v >> 2) & 1) ? -pv : pv;
        a2 += ((v >> 1) & 1) ? -pv : pv;
        a3 += ((v >> 0) & 1) ? -pv : pv;
    }
    float a0 = hi ? -sum0 : sum0;

    a0 += __shfl_xor(a0, 16, 32);
    a1 += __shfl_xor(a1, 16, 32);
    a2 += __shfl_xor(a2, 16, 32);
    a3 += __shfl_xor(a3, 16, 32);

    if (lane < 16) {
        float* o = out + (size_t)b * OUT_STRIDE_B + p;   // out[b][c][p]
        o[0]        = a0;
        o[1021]     = a1;
        o[2 * 1021] = a2;
        o[3 * 1021] = a3;
    }
}

// ---------------------------------------------------------------------------
extern "C" void kernel_launch(void* const* d_in, const int* in_sizes, int n_in,
                              void* d_out, int out_size, void* d_ws, size_t ws_size,
                              hipStream_t stream)
{
    const float* x      = (const float*)d_in[0];   // (512, 1024) f32
    const float* params = (const float*)d_in[1];   // (1021, 2, 4, 3) f32
    float*       out    = (float*)d_out;           // (512, 4, 1021) f32
    float*       U      = (float*)d_ws;            // 1021*512 floats = ~2 MB

    // Kernel 1: 1021*16 = 16336 threads
    build_unitaries_kernel<<<(N_POSI * 16 + 255) / 256, 256, 0, stream>>>(params, U);

    // Kernel 2: 1021 positions * 32 tiles = 32672 waves; 8 waves per block
    const int waves  = N_POSI * 32;
    const int blocks = waves / 8;                  // 4084 exactly
    qconv_wmma_kernel<<<blocks, 256, 0, stream>>>(x, U, out);
}